// GraphAttentionNet_68470368633045
// MI455X (gfx1250) — compile-verified
//
#include <hip/hip_runtime.h>
#include <hip/hip_bf16.h>
#include <math.h>

// ---------------------------------------------------------------------------
// GAT pipeline for gfx1250 (MI455X). DIM=128, HEADS=4, FH=32, LAYERS=3.
// Dense N x 128 x 128 GEMMs use V_WMMA_F32_16X16X4_F32 with W staged in LDS.
// Edge attention uses orderable-int atomicMax for segment_max and f32
// atomicAdd scatter (L2 atomics), with float4-vectorized loads everywhere.
// All intermediates live in d_ws.
// ---------------------------------------------------------------------------

typedef float v2f __attribute__((ext_vector_type(2)));
typedef float v8f __attribute__((ext_vector_type(8)));

#define DIM   128
#define HEADS 4
#define FH    32
#define NEG_SLOPE 0.2f
#define GAT_EPS 1e-16f

// monotonic float <-> int order mapping (works for negatives)
__device__ __forceinline__ int f2ord(float v) {
    int i = __float_as_int(v);
    return (i >= 0) ? i : (i ^ 0x7fffffff);
}
__device__ __forceinline__ float ord2f(int i) {
    return __int_as_float((i >= 0) ? i : (i ^ 0x7fffffff));
}
#define ORD_NEG_INF ((int)0x807FFFFFu)  // f2ord(-inf)

__device__ __forceinline__ float lrelu(float v) {
    return (v > 0.f) ? v : NEG_SLOPE * v;
}

// ---------------------------------------------------------------- fill utils
__global__ void fill_f32_kernel(float* p, float v, int n) {
    int i = blockIdx.x * blockDim.x + threadIdx.x;
    if (i < n) p[i] = v;
}
__global__ void fill_i32_kernel(int* p, int v, int n) {
    int i = blockIdx.x * blockDim.x + threadIdx.x;
    if (i < n) p[i] = v;
}

// ------------------------------------------------------------- embed gather
__global__ void gather_kernel(const int* __restrict__ x,
                              const float* __restrict__ embed,
                              float* __restrict__ h, int n) {
    // one thread per (node, dim-quad): coalesced float4 copies
    int i = blockIdx.x * blockDim.x + threadIdx.x;
    if (i >= n * (DIM / 4)) return;
    int node = i >> 5, q = i & 31;
    const float4* src = (const float4*)(embed + (size_t)x[node] * DIM);
    float4*       dst = (float4*)(h + (size_t)node * DIM);
    dst[q] = src[q];
}

// --------------------------------------------------- WMMA GEMM: hW = h @ W
// One wave handles 16 rows x 128 cols: 8 accumulator tiles of 16x16,
// K loop of 32 steps of V_WMMA_F32_16X16X4_F32. W (128x128 f32, 64KB) is
// staged in LDS by the whole workgroup (8 waves).
__global__ void __launch_bounds__(256)
gemm128_wmma_f32(const float* __restrict__ A, const float* __restrict__ W,
                 float* __restrict__ Out, int n) {
    __shared__ float Wl[DIM * DIM];

    const int tid = threadIdx.x;
    const float4* Wv  = (const float4*)W;
    float4*       Wlv = (float4*)Wl;
#pragma unroll
    for (int i = 0; i < 16; ++i) Wlv[tid + i * 256] = Wv[tid + i * 256];
    __syncthreads();

    const int wave    = tid >> 5;
    const int lane    = tid & 31;
    const int rowBase = (blockIdx.x * 8 + wave) * 16;
    if (rowBase >= n) return;  // wave-uniform exit, EXEC stays all-ones

    v8f acc[8] = {};

    const int lh = lane >> 4;   // 0 or 1 (selects K pair)
    const int ll = lane & 15;
    int arow = rowBase + ll;
    if (arow >= n) arow = n - 1;            // clamp loads; stores predicated
    const float* aptr = A + (size_t)arow * DIM;

#pragma unroll 4
    for (int k0 = 0; k0 < DIM; k0 += 4) {
        const int kk = k0 + lh * 2;
        v2f a;
        a.x = aptr[kk];
        a.y = aptr[kk + 1];
#pragma unroll
        for (int t = 0; t < 8; ++t) {
            const int col = t * 16 + ll;
            v2f b;
            b.x = Wl[kk * DIM + col];
            b.y = Wl[(kk + 1) * DIM + col];
            acc[t] = __builtin_amdgcn_wmma_f32_16x16x4_f32(
                false, a, false, b, (short)0, acc[t], false, false);
        }
    }

#pragma unroll
    for (int t = 0; t < 8; ++t) {
        const int col = t * 16 + ll;
#pragma unroll
        for (int v = 0; v < 8; ++v) {
            const int row = rowBase + v + 8 * lh;
            if (row < n) Out[(size_t)row * DIM + col] = acc[t][v];
        }
    }
}

// ----------------------------------------- per-(node,head) attention coeffs
__global__ void alpha_kernel(const float* __restrict__ hW,
                             const float* __restrict__ att_src_l,
                             const float* __restrict__ att_dst_l,
                             float* __restrict__ asrc, float* __restrict__ adst,
                             int n) {
    int idx = blockIdx.x * blockDim.x + threadIdx.x;
    if (idx >= n * HEADS) return;
    int node = idx >> 2, hd = idx & 3;
    const float4* p  = (const float4*)(hW + (size_t)node * DIM + hd * FH);
    const float4* a1 = (const float4*)(att_src_l + hd * FH);
    const float4* a2 = (const float4*)(att_dst_l + hd * FH);
    float s1 = 0.f, s2 = 0.f;
#pragma unroll
    for (int q = 0; q < FH / 4; ++q) {
        float4 v = p[q], w1 = a1[q], w2 = a2[q];
        s1 += v.x * w1.x + v.y * w1.y + v.z * w1.z + v.w * w1.w;
        s2 += v.x * w2.x + v.y * w2.y + v.z * w2.z + v.w * w2.w;
    }
    asrc[idx] = s1;
    adst[idx] = s2;
}

// -------------------------------------------------- edge pass 1: segment max
__global__ void edge_max_kernel(const int* __restrict__ src,
                                const int* __restrict__ dst, int E, int n,
                                const float* __restrict__ asrc,
                                const float* __restrict__ adst,
                                int* __restrict__ maxi) {
    int e = blockIdx.x * blockDim.x + threadIdx.x;
    int T = E + n;                 // self-loops appended
    if (e >= T) return;
    int s, d;
    if (e < E) { s = src[e]; d = dst[e]; } else { s = d = e - E; }
    float4 as = ((const float4*)asrc)[s];
    float4 ad = ((const float4*)adst)[d];
    int* mp = maxi + d * HEADS;
    atomicMax(&mp[0], f2ord(lrelu(as.x + ad.x)));
    atomicMax(&mp[1], f2ord(lrelu(as.y + ad.y)));
    atomicMax(&mp[2], f2ord(lrelu(as.z + ad.z)));
    atomicMax(&mp[3], f2ord(lrelu(as.w + ad.w)));
}

// ------------------------------------------------ edge pass 2: exp + denom
__global__ void edge_expsum_kernel(const int* __restrict__ src,
                                   const int* __restrict__ dst, int E, int n,
                                   const float* __restrict__ asrc,
                                   const float* __restrict__ adst,
                                   const int* __restrict__ maxi,
                                   float* __restrict__ denom) {
    int e = blockIdx.x * blockDim.x + threadIdx.x;
    int T = E + n;
    if (e >= T) return;
    int s, d;
    if (e < E) { s = src[e]; d = dst[e]; } else { s = d = e - E; }
    float4 as = ((const float4*)asrc)[s];
    float4 ad = ((const float4*)adst)[d];
    int4   mi = ((const int4*)maxi)[d];
    float* dp = denom + d * HEADS;
    atomicAdd(&dp[0], expf(lrelu(as.x + ad.x) - ord2f(mi.x)));
    atomicAdd(&dp[1], expf(lrelu(as.y + ad.y) - ord2f(mi.y)));
    atomicAdd(&dp[2], expf(lrelu(as.z + ad.z) - ord2f(mi.z)));
    atomicAdd(&dp[3], expf(lrelu(as.w + ad.w) - ord2f(mi.w)));
}

// ----------------------------------------- edge pass 3: scatter messages
// one thread per (edge, head): 32-float message row via 8x float4 loads,
// 32 independent global_atomic_add_f32 scatters
__global__ void edge_scatter_kernel(const int* __restrict__ src,
                                    const int* __restrict__ dst, int E, int n,
                                    const float* __restrict__ asrc,
                                    const float* __restrict__ adst,
                                    const int* __restrict__ maxi,
                                    const float* __restrict__ denom,
                                    const float* __restrict__ hW,
                                    float* __restrict__ agg) {
    int idx = blockIdx.x * blockDim.x + threadIdx.x;
    int T = E + n;
    if (idx >= T * HEADS) return;
    int e = idx >> 2, h = idx & 3;
    int s, d;
    if (e < E) { s = src[e]; d = dst[e]; } else { s = d = e - E; }
    float v     = lrelu(asrc[s * HEADS + h] + adst[d * HEADS + h]);
    float m     = ord2f(maxi[d * HEADS + h]);
    float den   = denom[d * HEADS + h];
    float alpha = expf(v - m) / (den + GAT_EPS);

    const float4* hs = (const float4*)(hW + (size_t)s * DIM + h * FH);
    float*        ad = agg + (size_t)d * DIM + h * FH;
    __builtin_prefetch(hs, 0, 1);   // global_prefetch_b8
#pragma unroll
    for (int q = 0; q < FH / 4; ++q) {
        float4 mv = hs[q];
        atomicAdd(&ad[q * 4 + 0], alpha * mv.x);
        atomicAdd(&ad[q * 4 + 1], alpha * mv.y);
        atomicAdd(&ad[q * 4 + 2], alpha * mv.z);
        atomicAdd(&ad[q * 4 + 3], alpha * mv.w);
    }
}

// --------------------------------------------------- relu(agg + bias) -> h
__global__ void finalize_kernel(const float* __restrict__ agg,
                                const float* __restrict__ bias_l,
                                float* __restrict__ h, int n) {
    int i = blockIdx.x * blockDim.x + threadIdx.x;
    if (i >= n * DIM) return;
    h[i] = fmaxf(agg[i] + bias_l[i & 127], 0.f);
}

// ------------------------------------------- global attention pooling pass
// phase 1: each thread -> softmax numerator of its node
// phase 2: first 128 lanes do coalesced weighted accumulation over the
//          block's rows, then one global atomic per (block, dim)
__global__ void __launch_bounds__(256)
pool_kernel(const float* __restrict__ x,
            const float* __restrict__ gate_w,
            const float* __restrict__ gate_b,
            float* __restrict__ gacc, float* __restrict__ gden, int n) {
    __shared__ float sex[256];
    const int t    = threadIdx.x;
    const int base = blockIdx.x * 256;
    const int i    = base + t;

    float ex = 0.f;
    if (i < n) {
        const float4* xi = (const float4*)(x + (size_t)i * DIM);
        const float4* gw = (const float4*)gate_w;
        float s = gate_b[0];
#pragma unroll 8
        for (int q = 0; q < DIM / 4; ++q) {
            float4 xv = xi[q], gv = gw[q];
            s += xv.x * gv.x + xv.y * gv.y + xv.z * gv.z + xv.w * gv.w;
        }
        float p = 1.f / (1.f + expf(-s));   // sigmoid
        ex = expf(p);                        // softmax numerator (p in (0,1))
    }
    sex[t] = ex;
    __syncthreads();

    const int limit = (n - base < 256) ? (n - base) : 256;
    if (t < DIM) {
        float acc = 0.f;
        for (int j = 0; j < limit; ++j)
            acc += sex[j] * x[(size_t)(base + j) * DIM + t];
        atomicAdd(&gacc[t], acc);
    } else if (t == DIM) {
        float s = 0.f;
        for (int j = 0; j < 256; ++j) s += sex[j];
        atomicAdd(gden, s);
    }
}

// ------------------------------------------------------------ MLP head
__global__ void head_kernel(const float* __restrict__ gacc,   // [2][128]
                            const float* __restrict__ gden,   // [2]
                            const float* __restrict__ fc1_w,  // [256][128]
                            const float* __restrict__ fc1_b,  // [128]
                            const float* __restrict__ fc2_w,  // [128]
                            const float* __restrict__ fc2_b,  // [1]
                            float* __restrict__ out) {
    __shared__ float cat[2 * DIM];
    __shared__ float red[DIM];
    int t = threadIdx.x;
    if (t < 2 * DIM) cat[t] = gacc[t] / gden[t >> 7];
    __syncthreads();
    if (t < DIM) {
        float s = fc1_b[t];
        for (int k = 0; k < 2 * DIM; ++k) s += cat[k] * fc1_w[k * DIM + t];
        red[t] = fmaxf(s, 0.f) * fc2_w[t];
    }
    __syncthreads();
    if (t == 0) {
        float s = fc2_b[0];
        for (int j = 0; j < DIM; ++j) s += red[j];
        out[0] = s;
    }
}

// ---------------------------------------------------------------------------
extern "C" void kernel_launch(void* const* d_in, const int* in_sizes, int n_in,
                              void* d_out, int out_size, void* d_ws, size_t ws_size,
                              hipStream_t stream) {
    const int*   x1     = (const int*)  d_in[0];
    const int*   x2     = (const int*)  d_in[1];
    const int*   ei1    = (const int*)  d_in[2];   // [2, E] flat
    const int*   ei2    = (const int*)  d_in[3];
    const float* embed  = (const float*)d_in[4];
    const float* W      = (const float*)d_in[5];   // [3,128,128]
    const float* attS   = (const float*)d_in[6];   // [3,4,32]
    const float* attD   = (const float*)d_in[7];
    const float* bias   = (const float*)d_in[8];   // [3,128]
    const float* gate_w = (const float*)d_in[9];   // [128]
    const float* gate_b = (const float*)d_in[10];  // [1]
    const float* fc1_w  = (const float*)d_in[11];  // [256,128]
    const float* fc1_b  = (const float*)d_in[12];  // [128]
    const float* fc2_w  = (const float*)d_in[13];  // [128]
    const float* fc2_b  = (const float*)d_in[14];  // [1]
    float*       out    = (float*)d_out;

    const int N = in_sizes[0];
    const int E = in_sizes[2] / 2;

    // workspace carve-up (floats); base is 256B-aligned from the harness
    float* ws    = (float*)d_ws;
    float* h     = ws;  ws += (size_t)N * DIM;
    float* hW    = ws;  ws += (size_t)N * DIM;
    float* agg   = ws;  ws += (size_t)N * DIM;
    float* asrc  = ws;  ws += (size_t)N * HEADS;
    float* adst  = ws;  ws += (size_t)N * HEADS;
    float* denom = ws;  ws += (size_t)N * HEADS;
    int*   maxi  = (int*)ws; ws += (size_t)N * HEADS;
    float* gacc  = ws;  ws += 2 * DIM;
    float* gden  = ws;  ws += 2;

    auto cdiv = [](int a, int b) { return (a + b - 1) / b; };
    const int BT = 256;
    const int T  = E + N;   // edges incl. self loops

    // zero pooling accumulators (harness does not re-poison between replays)
    fill_f32_kernel<<<1, 2 * DIM + 2, 0, stream>>>(gacc, 0.f, 2 * DIM + 2);

    for (int g = 0; g < 2; ++g) {
        const int* x   = (g == 0) ? x1  : x2;
        const int* src = (g == 0) ? ei1 : ei2;
        const int* dst = src + E;

        gather_kernel<<<cdiv(N * (DIM / 4), BT), BT, 0, stream>>>(x, embed, h, N);

        for (int l = 0; l < 3; ++l) {
            const float* Wl = W    + (size_t)l * DIM * DIM;
            const float* aS = attS + (size_t)l * HEADS * FH;
            const float* aD = attD + (size_t)l * HEADS * FH;
            const float* bl = bias + (size_t)l * DIM;

            gemm128_wmma_f32<<<cdiv(N, 16 * 8), BT, 0, stream>>>(h, Wl, hW, N);
            alpha_kernel<<<cdiv(N * HEADS, BT), BT, 0, stream>>>(hW, aS, aD,
                                                                 asrc, adst, N);
            fill_i32_kernel<<<cdiv(N * HEADS, BT), BT, 0, stream>>>(
                maxi, ORD_NEG_INF, N * HEADS);
            fill_f32_kernel<<<cdiv(N * HEADS, BT), BT, 0, stream>>>(
                denom, 0.f, N * HEADS);
            fill_f32_kernel<<<cdiv(N * DIM, BT), BT, 0, stream>>>(
                agg, 0.f, N * DIM);

            edge_max_kernel<<<cdiv(T, BT), BT, 0, stream>>>(src, dst, E, N,
                                                            asrc, adst, maxi);
            edge_expsum_kernel<<<cdiv(T, BT), BT, 0, stream>>>(
                src, dst, E, N, asrc, adst, maxi, denom);
            edge_scatter_kernel<<<cdiv(T * HEADS, BT), BT, 0, stream>>>(
                src, dst, E, N, asrc, adst, maxi, denom, hW, agg);

            finalize_kernel<<<cdiv(N * DIM, BT), BT, 0, stream>>>(agg, bl, h, N);
        }

        pool_kernel<<<cdiv(N, BT), BT, 0, stream>>>(h, gate_w, gate_b,
                                                    gacc + g * DIM, gden + g, N);
    }

    head_kernel<<<1, 256, 0, stream>>>(gacc, gden, fc1_w, fc1_b, fc2_w, fc2_b,
                                       out);
}